// PerPixelChannelPermutation_11974368821258
// MI455X (gfx1250) — compile-verified
//
#include <hip/hip_runtime.h>

// out[i,j,k] = image[i,j, perm[i,j,k]]  for image, perm of shape (4096, 4096, 3)
//
// Pure streaming gather: 0 FLOPs, ~576 MiB of HBM traffic per call.
// Bound by 23.3 TB/s HBM => ~26 us floor. Strategy: everything as b128
// non-temporal vector memory ops, gather resolved in registers with
// branch-free v_cndmask selects (perm values are only 0/1/2).

typedef __attribute__((ext_vector_type(4))) float f32x4;
typedef __attribute__((ext_vector_type(4))) int   i32x4;

#define NPIXELS   (4096LL * 4096LL)          // 16,777,216 pixels
#define NELEMS    (NPIXELS * 3LL)            // 50,331,648 floats (divisible by 12)
#define BLOCK     256
#define NTHREADS  (NELEMS / 12LL)            // 4 pixels (3 float4s) per thread
#define NBLOCKS   ((int)(NTHREADS / BLOCK))  // 16384

// Branch-free 3-way select: p in {0,1,2}
__device__ __forceinline__ float pick3(float c0, float c1, float c2, int p) {
  float r = (p == 1) ? c1 : c0;
  return (p == 2) ? c2 : r;
}

__global__ __launch_bounds__(BLOCK)
void PerPixelChannelPermutation_kernel(const float* __restrict__ img,
                                       const int*   __restrict__ prm,
                                       float*       __restrict__ out) {
  const long long t = (long long)blockIdx.x * BLOCK + threadIdx.x;
  const long long b = t * 3;  // index in units of float4 / int4

  const f32x4* __restrict__ img4 = (const f32x4*)img;
  const i32x4* __restrict__ prm4 = (const i32x4*)prm;
  f32x4*       __restrict__ out4 = (f32x4*)out;

  // 3 x b128 NT loads of image values (12 floats = 4 pixels)
  f32x4 v0 = __builtin_nontemporal_load(img4 + b + 0);
  f32x4 v1 = __builtin_nontemporal_load(img4 + b + 1);
  f32x4 v2 = __builtin_nontemporal_load(img4 + b + 2);
  // 3 x b128 NT loads of permutation indices
  i32x4 p0 = __builtin_nontemporal_load(prm4 + b + 0);
  i32x4 p1 = __builtin_nontemporal_load(prm4 + b + 1);
  i32x4 p2 = __builtin_nontemporal_load(prm4 + b + 2);

  f32x4 o0, o1, o2;
  // pixel 0: channels (v0.x, v0.y, v0.z), perms (p0.x, p0.y, p0.z)
  o0.x = pick3(v0.x, v0.y, v0.z, p0.x);
  o0.y = pick3(v0.x, v0.y, v0.z, p0.y);
  o0.z = pick3(v0.x, v0.y, v0.z, p0.z);
  // pixel 1: channels (v0.w, v1.x, v1.y), perms (p0.w, p1.x, p1.y)
  o0.w = pick3(v0.w, v1.x, v1.y, p0.w);
  o1.x = pick3(v0.w, v1.x, v1.y, p1.x);
  o1.y = pick3(v0.w, v1.x, v1.y, p1.y);
  // pixel 2: channels (v1.z, v1.w, v2.x), perms (p1.z, p1.w, p2.x)
  o1.z = pick3(v1.z, v1.w, v2.x, p1.z);
  o1.w = pick3(v1.z, v1.w, v2.x, p1.w);
  o2.x = pick3(v1.z, v1.w, v2.x, p2.x);
  // pixel 3: channels (v2.y, v2.z, v2.w), perms (p2.y, p2.z, p2.w)
  o2.y = pick3(v2.y, v2.z, v2.w, p2.y);
  o2.z = pick3(v2.y, v2.z, v2.w, p2.z);
  o2.w = pick3(v2.y, v2.z, v2.w, p2.w);

  // 3 x b128 NT stores
  __builtin_nontemporal_store(o0, out4 + b + 0);
  __builtin_nontemporal_store(o1, out4 + b + 1);
  __builtin_nontemporal_store(o2, out4 + b + 2);
}

extern "C" void kernel_launch(void* const* d_in, const int* in_sizes, int n_in,
                              void* d_out, int out_size, void* d_ws, size_t ws_size,
                              hipStream_t stream) {
  const float* img = (const float*)d_in[0];  // image, float32, 4096*4096*3
  const int*   prm = (const int*)d_in[1];    // perm,  int32,   4096*4096*3
  float*       out = (float*)d_out;          // float32, 4096*4096*3

  PerPixelChannelPermutation_kernel<<<NBLOCKS, BLOCK, 0, stream>>>(img, prm, out);
}